// CTCLayer_54494545051874
// MI455X (gfx1250) — compile-verified
//
#include <hip/hip_runtime.h>
#include <math.h>

// CTC forward scan for B=64, T=1024, L=128, V=512 (S = 2L+1 = 257).
// Linear-probability-domain scaled forward algorithm (no per-step transcendentals),
// one wave32 per batch row, 9 states per lane, TDM double-buffered row streaming.

#define KR 8          // time rows per TDM chunk (8 x 512 floats = 16KB per chunk)
#define PSTATES 9     // states per lane: 32*9 = 288 >= S = 257

typedef unsigned int u32x4 __attribute__((ext_vector_type(4)));
typedef int          i32x8 __attribute__((ext_vector_type(8)));
typedef int          i32x4 __attribute__((ext_vector_type(4)));

// Issue a TDM load of [nrows x rowlen] contiguous f32 rows into LDS.
// D# layout per CDNA5 ISA.
__device__ __forceinline__ void tdm_load_rows(const float* gsrc,
                                              unsigned lds_byte_off,
                                              int nrows, int rowlen) {
  unsigned long long ga = (unsigned long long)(size_t)gsrc;
  u32x4 g0;
  g0[0] = 1u;                                                  // count=1 (valid user D#)
  g0[1] = lds_byte_off;                                        // lds_addr (bytes)
  g0[2] = (unsigned)(ga & 0xffffffffull);                      // global_addr[31:0]
  g0[3] = (unsigned)((ga >> 32) & 0x1ffffffull) | (2u << 30);  // global_addr[56:32] | type=2

  i32x8 g1;
  g1[0] = (int)(2u << 16);                                     // data_size=2 (4 bytes), mask=0
  g1[1] = (int)(((unsigned)rowlen & 0xffffu) << 16);           // tensor_dim0[15:0]
  g1[2] = (int)((((unsigned)rowlen >> 16) & 0xffffu) |
                (((unsigned)nrows & 0xffffu) << 16));          // td0[31:16] | tensor_dim1[15:0]
  g1[3] = (int)(((unsigned)rowlen & 0xffffu) << 16);           // td1[31:16]=0 | tile_dim0
  g1[4] = (int)((unsigned)nrows & 0xffffu);                    // tile_dim1 | tile_dim2=0
  g1[5] = rowlen;                                              // tensor_dim0_stride[31:0]
  g1[6] = 0;                                                   // stride0[47:32] | stride1[15:0]
  g1[7] = 0;                                                   // stride1[47:16]

  i32x4 z4 = {0, 0, 0, 0};                                     // groups 2/3 unused (tile_dim2=0)
  i32x8 z8 = {0, 0, 0, 0, 0, 0, 0, 0};                         // trailing group (6-arg variant)
  __builtin_amdgcn_tensor_load_to_lds(g0, g1, z4, z4, z8, 0);
}

__global__ __launch_bounds__(32) void ctc_scan_kernel(
    const int* __restrict__ y_true,   // [B, L] labels in [0, V-2]
    const float* __restrict__ y_pred, // [B, T, V] softmax probabilities
    float* __restrict__ out,          // [B] loss
    int B, int T, int L, int V) {
  const int b     = blockIdx.x;
  const int lane  = threadIdx.x;      // wave32
  const int S     = 2 * L + 1;
  const int blank = V - 1;

  __shared__ float rows[2][KR][512];  // V = 512 row length, double buffered

  // ---- per-state constants (time-invariant) ----
  int   off[PSTATES];                 // vocab gather index for this state
  float mskip[PSTATES];               // allow_skip mask (1.0 / 0.0)
  float pmask[PSTATES];               // 1.0 for valid states (s < S), else 0.0
  float peps[PSTATES];                // EPS for valid states, else 0.0 (keeps dead states at 0)
  float a[PSTATES];                   // alpha (linear domain)

#pragma unroll
  for (int j = 0; j < PSTATES; ++j) {
    int s = lane * PSTATES + j;
    bool valid = (s < S);
    int   e  = blank;
    float ms = 0.f;
    if (valid && (s & 1)) {           // odd extended state = label
      int li = s >> 1;
      e = y_true[b * L + li];
      int prev = (li == 0) ? -1 : y_true[b * L + li - 1];
      ms = (e != prev && e != blank) ? 1.f : 0.f;
    }
    off[j]   = valid ? e : 0;
    mskip[j] = ms;
    pmask[j] = valid ? 1.f : 0.f;
    peps[j]  = valid ? 1e-7f : 0.f;
    a[j]     = 0.f;
  }
  // virtual alpha_{-1}[0] = 1 makes the uniform step at t=0 produce alpha0 exactly
  if (lane == 0) a[0] = 1.f;
  int etotal = 0;                     // accumulated base-2 exponent of the scale

  const float* yb  = y_pred + (size_t)b * T * V;
  const int    nch = T / KR;

  // ---- prologue: two TDM chunks in flight ----
  tdm_load_rows(yb, (unsigned)(size_t)&rows[0][0][0], KR, V);
  if (nch > 1)
    tdm_load_rows(yb + (size_t)KR * V, (unsigned)(size_t)&rows[1][0][0], KR, V);

  for (int c = 0; c < nch; ++c) {
    // TENSORcnt is in-order per wave: <=1 outstanding means the current chunk landed.
    if (c + 1 < nch) __builtin_amdgcn_s_wait_tensorcnt(1);
    else             __builtin_amdgcn_s_wait_tensorcnt(0);

    const float* base = &rows[c & 1][0][0];
#pragma unroll 1
    for (int r = 0; r < KR; ++r) {
      const float* rp = base + r * 512;

      // previous lane's top two states (alpha[s-1], alpha[s-2] across the lane seam)
      float t1 = __shfl_up(a[PSTATES - 1], 1, 32);
      float t2 = __shfl_up(a[PSTATES - 2], 1, 32);
      if (lane == 0) { t1 = 0.f; t2 = 0.f; }   // linear-domain "NEG" is exactly 0

      float na[PSTATES];
#pragma unroll
      for (int j = 0; j < PSTATES; ++j) {
        float am1 = (j == 0) ? t1 : a[j - 1];
        float am2 = (j == 0) ? t2 : ((j == 1) ? t1 : a[j - 2]);
        float sum = fmaf(mskip[j], am2, a[j] + am1);
        float pv  = rp[off[j]];                         // LDS gather (blank = broadcast)
        na[j] = sum * fmaf(pv, pmask[j], peps[j]);      // * (p + EPS), dead states -> 0
      }
#pragma unroll
      for (int j = 0; j < PSTATES; ++j) a[j] = na[j];

      // exact power-of-two rescale every 4 steps (p >= 1e-7 => max decays <= 2^-93 between)
      if ((r & 3) == 3) {
        float m = a[0];
#pragma unroll
        for (int j = 1; j < PSTATES; ++j) m = fmaxf(m, a[j]);
        for (int o = 16; o > 0; o >>= 1) m = fmaxf(m, __shfl_xor(m, o, 32));
        if (m > 0.f) {
          int e;
          (void)frexpf(m, &e);                 // m = mant * 2^e, mant in [0.5, 1)
          etotal += e;
#pragma unroll
          for (int j = 0; j < PSTATES; ++j) a[j] = ldexpf(a[j], -e);
        }
      }
    }

    // refill the buffer we just consumed with chunk c+2
    if (c + 2 < nch)
      tdm_load_rows(yb + (size_t)(c + 2) * KR * V,
                    (unsigned)(size_t)&rows[c & 1][0][0], KR, V);
  }

  // total prob ends in last blank (S-1) or last label (S-2)
  float val = 0.f;
#pragma unroll
  for (int j = 0; j < PSTATES; ++j) {
    int s = lane * PSTATES + j;
    if (s == S - 1 || s == S - 2) val += a[j];
  }
  for (int o = 16; o > 0; o >>= 1) val += __shfl_xor(val, o, 32);

  if (lane == 0)
    out[b] = -(logf(val) + (float)etotal * 0.69314718055994530942f);
}

extern "C" void kernel_launch(void* const* d_in, const int* in_sizes, int n_in,
                              void* d_out, int out_size, void* d_ws, size_t ws_size,
                              hipStream_t stream) {
  const int*   y_true = (const int*)d_in[0];   // [B, L]
  const float* y_pred = (const float*)d_in[1]; // [B, T, V] float32
  float*       out    = (float*)d_out;         // [B, 1] float32

  const int B = out_size;              // reference returns [B, 1]
  const int L = in_sizes[0] / B;       // 128
  const int V = 512;                   // vocab (fixed by problem / LDS row size)
  const int T = in_sizes[1] / (B * V); // 1024

  ctc_scan_kernel<<<dim3(B), dim3(32), 0, stream>>>(y_true, y_pred, out, B, T, L, V);
}